// Model_42064909697613
// MI455X (gfx1250) — compile-verified
//
#include <hip/hip_runtime.h>
#include <hip/hip_bf16.h>
#include <cstddef>

// EGNN diffusion step, CDNA5 (gfx1250, wave32).
// Messages are masked to chain adjacency (|i-j|==1) and pos_mlp(0)=c0 is a
// constant, so the N^2 pair sum collapses to
//   c0*(S - N*x_i) + sum_{j in {i-1,i+1}} (pos_mlp(m_ij)-c0)*(x_j-x_i).
// All 64-wide MLP layers run on V_WMMA_F32_16X16X4_F32 (full f32 precision).
// Weights are pre-swizzled into B-fragment layout (zero-padded K) so the GEMM
// inner loop is branch-free: 1 A load (b64) + 4 B loads (b64) + 4 WMMAs.

typedef float v2f __attribute__((ext_vector_type(2)));
typedef float v8f __attribute__((ext_vector_type(8)));

#define Bn 2
#define Nn 512
#define NNODE 1024   /* Bn*Nn */
#define NPAIR 2048   /* Bn*2*Nn (2 slots per node: j=i-1, j=i+1) */

// ---------------------------------------------------------------------------
// Weight swizzle: W[kreal x 64] -> Wz[4 ntiles][kpad/4][32 lanes][2]
// matching the V_WMMA_F32_16X16X4_F32 B-fragment layout
// (VGPR0: K=k+0 lanes 0-15 / K=k+2 lanes 16-31; VGPR1: K=k+1 / K=k+3).
// Rows >= kreal are zero (K padding).
// ---------------------------------------------------------------------------
__global__ void k_swzW(const float* __restrict__ W, int kreal, int kpad,
                       float* __restrict__ Wz)
{
    const int idx   = blockIdx.x * blockDim.x + threadIdx.x;
    const int kb4   = kpad >> 2;
    const int total = kb4 * 128;            // 4 ntiles * kb4 * 32 lanes
    if (idx >= total) return;
    const int lane = idx & 31;
    const int kb   = (idx >> 5) % kb4;
    const int nt   = (idx >> 5) / kb4;
    const int half = lane >> 4, l15 = lane & 15;
    const int r0   = kb * 4 + half * 2;
    const int col  = nt * 16 + l15;
    const float v0 = (r0     < kreal) ? W[(size_t)r0 * 64 + col]       : 0.0f;
    const float v1 = (r0 + 1 < kreal) ? W[(size_t)(r0 + 1) * 64 + col] : 0.0f;
    const size_t o = ((size_t)(nt * kb4 + kb) * 32 + lane) * 2;
    Wz[o] = v0; Wz[o + 1] = v1;
}

// ---------------------------------------------------------------------------
// f32 WMMA GEMM: Y[M x 64] = act(X[M x kpad] @ W[.. x 64] + bias)
// grid.x = M/16, block = 32 (one wave, EXEC all ones, branch-free body).
// One wave computes the full 16x64 output row-block: 4 independent
// accumulator chains hide the WMMA D->C RAW latency; A fragment reused 4x.
// A fragment (16x4 f32): lanes 0-15 rows, VGPR0 K=0(lo)/K=2(hi), VGPR1 K=1/K=3.
// ---------------------------------------------------------------------------
__global__ __launch_bounds__(32)
void k_gemm64(const float* __restrict__ X, int ldx, int kpad,
              const float* __restrict__ Wz, const float* __restrict__ bias,
              float* __restrict__ Y, int relu)
{
    const int m0   = blockIdx.x << 4;
    const int lane = threadIdx.x;
    const int half = lane >> 4;     // 0: lanes 0-15, 1: lanes 16-31
    const int l15  = lane & 15;
    const int kb4  = kpad >> 2;

    const float* xr = X + (size_t)(m0 + l15) * ldx + (half << 1);
    const v2f*   wp = (const v2f*)Wz + lane;

    v8f acc0 = {0.f,0.f,0.f,0.f,0.f,0.f,0.f,0.f};
    v8f acc1 = acc0, acc2 = acc0, acc3 = acc0;

    for (int kb = 0; kb < kb4; ++kb) {
        v2f a;
        a.x = xr[kb * 4];
        a.y = xr[kb * 4 + 1];
        const v2f b0 = wp[(size_t)(0 * kb4 + kb) * 32];
        const v2f b1 = wp[(size_t)(1 * kb4 + kb) * 32];
        const v2f b2 = wp[(size_t)(2 * kb4 + kb) * 32];
        const v2f b3 = wp[(size_t)(3 * kb4 + kb) * 32];
        acc0 = __builtin_amdgcn_wmma_f32_16x16x4_f32(false, a, false, b0, (short)0, acc0, false, false);
        acc1 = __builtin_amdgcn_wmma_f32_16x16x4_f32(false, a, false, b1, (short)0, acc1, false, false);
        acc2 = __builtin_amdgcn_wmma_f32_16x16x4_f32(false, a, false, b2, (short)0, acc2, false, false);
        acc3 = __builtin_amdgcn_wmma_f32_16x16x4_f32(false, a, false, b3, (short)0, acc3, false, false);
    }

    union { v8f v; float f[8]; } u[4];
    u[0].v = acc0; u[1].v = acc1; u[2].v = acc2; u[3].v = acc3;
#pragma unroll
    for (int nt = 0; nt < 4; ++nt) {
        const float bv = bias[nt * 16 + l15];
#pragma unroll
        for (int v = 0; v < 8; ++v) {
            float val = u[nt].f[v] + bv;
            if (relu) val = fmaxf(val, 0.0f);
            // C/D layout: VGPR v -> row v (lanes 0-15) / row v+8 (lanes 16-31)
            Y[(size_t)(m0 + v + (half << 3)) * 64 + nt * 16 + l15] = val;
        }
    }
}

// ---------------------------------------------------------------------------
// h1[b,i,0:36] = [features(16), posenc(16), positions(3), t/1000]
// ---------------------------------------------------------------------------
__global__ void k_prep(const float* __restrict__ pos, const float* __restrict__ feat,
                       const int* __restrict__ t, float* __restrict__ h1)
{
    int idx = blockIdx.x * blockDim.x + threadIdx.x;
    if (idx >= NNODE) return;
    const int i = idx & (Nn - 1);
    float* h = h1 + (size_t)idx * 36;
    const float* f = feat + (size_t)idx * 16;
    for (int c = 0; c < 16; ++c) h[c] = f[c];
    const float fi = (float)i;
#pragma unroll
    for (int k = 0; k < 8; ++k) {
        float div = expf((float)(2 * k) * (-9.210340371976184f / 16.0f)); // -ln(1e4)/d
        h[16 + 2 * k]     = sinf(fi * div);
        h[16 + 2 * k + 1] = cosf(fi * div);
    }
    const float* x = pos + (size_t)idx * 3;
    h[32] = x[0]; h[33] = x[1]; h[34] = x[2];
    h[35] = (float)t[0] * (1.0f / 1000.0f);
}

// ---------------------------------------------------------------------------
// scal layout: [0..5]=S1 (per-batch sums of x), [8..10]=c0 layer1,
//              [12..14]=c0 layer2, [16..21]=S2 (per-batch sums of z1)
// Deterministic serial reductions (no float atomics).
// ---------------------------------------------------------------------------
__global__ void k_scalars1(const float* __restrict__ pos,
                           const float* __restrict__ p1b0, const float* __restrict__ p1W1,
                           const float* __restrict__ p1b1, const float* __restrict__ p1W2,
                           const float* __restrict__ p1b2,
                           const float* __restrict__ p2b0, const float* __restrict__ p2W1,
                           const float* __restrict__ p2b1, const float* __restrict__ p2W2,
                           const float* __restrict__ p2b2,
                           float* __restrict__ scal)
{
    const int t = threadIdx.x;
    if (t < 6) {
        const int b = t / 3, c = t % 3;
        float s = 0.f;
        for (int i = 0; i < Nn; ++i) s += pos[(size_t)(b * Nn + i) * 3 + c];
        scal[t] = s;
    } else if (t == 6 || t == 7) {
        // c0 = pos_mlp(0): layer0 reduces to bias, then two dense layers.
        const float* b0 = (t == 6) ? p1b0 : p2b0;
        const float* W1 = (t == 6) ? p1W1 : p2W1;
        const float* b1 = (t == 6) ? p1b1 : p2b1;
        const float* W2 = (t == 6) ? p1W2 : p2W2;
        const float* b2 = (t == 6) ? p1b2 : p2b2;
        float x1[64], x2[64];
        for (int k = 0; k < 64; ++k) x1[k] = fmaxf(b0[k], 0.f);
        for (int j = 0; j < 64; ++j) {
            float s = b1[j];
            for (int k = 0; k < 64; ++k) s += x1[k] * W1[k * 64 + j];
            x2[j] = fmaxf(s, 0.f);
        }
        float* out = scal + ((t == 6) ? 8 : 12);
        for (int c = 0; c < 3; ++c) {
            float s = b2[c];
            for (int k = 0; k < 64; ++k) s += x2[k] * W2[k * 3 + c];
            out[c] = s;
        }
    }
}

__global__ void k_scalars2(const float* __restrict__ z1, float* __restrict__ scal)
{
    const int t = threadIdx.x;
    if (t < 6) {
        const int b = t / 3, c = t % 3;
        float s = 0.f;
        for (int i = 0; i < Nn; ++i) s += z1[(size_t)(b * Nn + i) * 3 + c];
        scal[16 + t] = s;
    }
}

// ---------------------------------------------------------------------------
// Pair feature builders. Pair p = b*1024 + 2*i + s; s=0 -> j=i-1, s=1 -> j=i+1.
// Invalid edge slots get zero rows (their outputs are never consumed).
// ---------------------------------------------------------------------------
__global__ void k_buildF1(const float* __restrict__ h1, const float* __restrict__ pos,
                          float* __restrict__ F1)
{
    int p = blockIdx.x * blockDim.x + threadIdx.x;
    if (p >= NPAIR) return;
    const int b = p >> 10, r = p & 1023, i = r >> 1, s = r & 1;
    const int j = s ? i + 1 : i - 1;
    float* f = F1 + (size_t)p * 76;
    const bool valid = s ? (i < Nn - 1) : (i > 0);
    if (!valid) { for (int c = 0; c < 76; ++c) f[c] = 0.f; return; }
    const float* hj = h1 + (size_t)(b * Nn + j) * 36;
    const float* hi = h1 + (size_t)(b * Nn + i) * 36;
    for (int c = 0; c < 36; ++c) f[c]      = hj[c];
    for (int c = 0; c < 36; ++c) f[36 + c] = hi[c];
    const float* xi = pos + (size_t)(b * Nn + i) * 3;
    const float* xj = pos + (size_t)(b * Nn + j) * 3;
    const float dx = xj[0] - xi[0], dy = xj[1] - xi[1], dz = xj[2] - xi[2];
    f[72] = dx * dx + dy * dy + dz * dz;
    f[73] = 0.f; f[74] = 0.f; f[75] = 0.f;
}

__global__ void k_buildF2(const float* __restrict__ h2, const float* __restrict__ z1,
                          float* __restrict__ F2)
{
    int p = blockIdx.x * blockDim.x + threadIdx.x;
    if (p >= NPAIR) return;
    const int b = p >> 10, r = p & 1023, i = r >> 1, s = r & 1;
    const int j = s ? i + 1 : i - 1;
    float* f = F2 + (size_t)p * 132;
    const bool valid = s ? (i < Nn - 1) : (i > 0);
    if (!valid) { for (int c = 0; c < 132; ++c) f[c] = 0.f; return; }
    const float* hj = h2 + (size_t)(b * Nn + j) * 64;
    const float* hi = h2 + (size_t)(b * Nn + i) * 64;
    for (int c = 0; c < 64; ++c) f[c]      = hj[c];
    for (int c = 0; c < 64; ++c) f[64 + c] = hi[c];
    const float* xi = z1 + (size_t)(b * Nn + i) * 3;
    const float* xj = z1 + (size_t)(b * Nn + j) * 3;
    const float dx = xj[0] - xi[0], dy = xj[1] - xi[1], dz = xj[2] - xi[2];
    f[128] = dx * dx + dy * dy + dz * dz;
    f[129] = 0.f; f[130] = 0.f; f[131] = 0.f;
}

// pos-MLP output head: 64 -> 3, stored with stride 4.
__global__ void k_pos_last(const float* __restrict__ H, const float* __restrict__ W,
                           const float* __restrict__ bias, float* __restrict__ pout)
{
    int p = blockIdx.x * blockDim.x + threadIdx.x;
    if (p >= NPAIR) return;
    const float* h = H + (size_t)p * 64;
    float o0 = bias[0], o1 = bias[1], o2 = bias[2];
    for (int k = 0; k < 64; ++k) {
        const float hv = h[k];
        o0 += hv * W[k * 3 + 0];
        o1 += hv * W[k * 3 + 1];
        o2 += hv * W[k * 3 + 2];
    }
    pout[p * 4 + 0] = o0; pout[p * 4 + 1] = o1; pout[p * 4 + 2] = o2;
}

// feat-MLP input: [h1(36), msum(64)] where msum = sum of the <=2 neighbor msgs.
__global__ void k_featin(const float* __restrict__ h1, const float* __restrict__ m1,
                         float* __restrict__ Ff)
{
    int idx = blockIdx.x * blockDim.x + threadIdx.x;
    if (idx >= NNODE) return;
    const int b = idx >> 9, i = idx & (Nn - 1);
    float* f = Ff + (size_t)idx * 100;
    const float* h = h1 + (size_t)idx * 36;
    for (int c = 0; c < 36; ++c) f[c] = h[c];
    const float* mA = m1 + (size_t)(b * 1024 + i * 2 + 0) * 64; // j = i-1
    const float* mB = m1 + (size_t)(b * 1024 + i * 2 + 1) * 64; // j = i+1
    const bool vA = (i > 0), vB = (i < Nn - 1);
    for (int c = 0; c < 64; ++c) {
        float s = 0.f;
        if (vA) s += mA[c];
        if (vB) s += mB[c];
        f[36 + c] = s;
    }
}

// x-update: x + (c0*(S - N*x_i) + sum_nb (pos(m)-c0)*(x_j-x_i)) / (N-1), then relu.
__global__ void k_xupd1(const float* __restrict__ pos, const float* __restrict__ pout,
                        const float* __restrict__ scal, float* __restrict__ z1)
{
    int idx = blockIdx.x * blockDim.x + threadIdx.x;
    if (idx >= NNODE) return;
    const int b = idx >> 9, i = idx & (Nn - 1);
    const float* xi = pos + (size_t)idx * 3;
    const float* c0 = scal + 8;
    const float* S  = scal + b * 3;
    float acc[3];
    for (int c = 0; c < 3; ++c) acc[c] = c0[c] * (S[c] - (float)Nn * xi[c]);
    for (int s = 0; s < 2; ++s) {
        const bool valid = s ? (i < Nn - 1) : (i > 0);
        if (!valid) continue;
        const int j = s ? i + 1 : i - 1;
        const int p = b * 1024 + i * 2 + s;
        const float* xj = pos + (size_t)(b * Nn + j) * 3;
        for (int c = 0; c < 3; ++c)
            acc[c] += (pout[p * 4 + c] - c0[c]) * (xj[c] - xi[c]);
    }
    for (int c = 0; c < 3; ++c)
        z1[(size_t)idx * 3 + c] = fmaxf(xi[c] + acc[c] * (1.0f / (float)(Nn - 1)), 0.0f);
}

// Final x-update (layer 2): no relu, writes d_out.
__global__ void k_final(const float* __restrict__ z1, const float* __restrict__ pout,
                        const float* __restrict__ scal, float* __restrict__ out)
{
    int idx = blockIdx.x * blockDim.x + threadIdx.x;
    if (idx >= NNODE) return;
    const int b = idx >> 9, i = idx & (Nn - 1);
    const float* xi = z1 + (size_t)idx * 3;
    const float* c0 = scal + 12;
    const float* S  = scal + 16 + b * 3;
    float acc[3];
    for (int c = 0; c < 3; ++c) acc[c] = c0[c] * (S[c] - (float)Nn * xi[c]);
    for (int s = 0; s < 2; ++s) {
        const bool valid = s ? (i < Nn - 1) : (i > 0);
        if (!valid) continue;
        const int j = s ? i + 1 : i - 1;
        const int p = b * 1024 + i * 2 + s;
        const float* xj = z1 + (size_t)(b * Nn + j) * 3;
        for (int c = 0; c < 3; ++c)
            acc[c] += (pout[p * 4 + c] - c0[c]) * (xj[c] - xi[c]);
    }
    for (int c = 0; c < 3; ++c)
        out[(size_t)idx * 3 + c] = xi[c] + acc[c] * (1.0f / (float)(Nn - 1));
}

// ---------------------------------------------------------------------------
extern "C" void kernel_launch(void* const* d_in, const int* in_sizes, int n_in,
                              void* d_out, int out_size, void* d_ws, size_t ws_size,
                              hipStream_t stream)
{
    const float* positions = (const float*)d_in[0];
    const float* features  = (const float*)d_in[1];
    const int*   t_ptr     = (const int*)d_in[n_in - 1];

    // Locate param groups by unique flat sizes (robust to pytree flatten order).
    // Each MLP contributes 6 consecutive leaves: W0,b0,W1,b1,W2,b2.
    int msg1 = -1, feat1 = -1, msg2 = -1, pos1 = -1, pos2 = -1;
    for (int i = 0; i < n_in; ++i) {
        const int s = in_sizes[i];
        if (s == 4672)      msg1 = i;                    // egnn1.msg W0: 73x64
        else if (s == 6400) feat1 = i;                   // egnn1.feat W0: 100x64
        else if (s == 8256) msg2 = i;                    // egnn2.msg W0: 129x64
        else if (s == 192 && i + 1 < n_in && in_sizes[i + 1] == 3) {
            if (pos1 < 0) pos1 = i - 4; else pos2 = i - 4; // pos W2: 64x3, b2: 3
        }
    }
    auto F = [&](int i) { return (const float*)d_in[i]; };

    // Workspace carve-up (floats).
    float* ws    = (float*)d_ws;
    float* h1    = ws;                    // 1024*36           = 36864
    float* Fbuf  = h1 + 36864;            // max(2048*76,2048*132) = 270336
    float* buf0  = Fbuf + 270336;         // 2048*64           = 131072
    float* buf1  = buf0 + 131072;
    float* buf2  = buf1 + 131072;
    float* pout  = buf2 + 131072;         // 2048*4            = 8192
    float* Ffeat = pout + 8192;           // 1024*100          = 102400
    float* h2    = Ffeat + 102400;        // 1024*64           = 65536
    float* z1    = h2 + 65536;            // 1024*3            = 3072
    float* scal  = z1 + 3072;             // 32
    float* wzbase= scal + 32;             // swizzled weights, ~60672 floats

    // Swizzled-weight table: {src leaf, kreal, kpad}
    const float* wsrc[13] = { F(msg1),  F(msg1 + 2), F(msg1 + 4),
                              F(pos1),  F(pos1 + 2),
                              F(feat1), F(feat1 + 2), F(feat1 + 4),
                              F(msg2),  F(msg2 + 2),  F(msg2 + 4),
                              F(pos2),  F(pos2 + 2) };
    const int kreal[13] = { 73, 64, 64,  64, 64,  100, 64, 64,  129, 64, 64,  64, 64 };
    const int kpadv[13] = { 76, 64, 64,  64, 64,  100, 64, 64,  132, 64, 64,  64, 64 };
    float* wz[13];
    {
        float* p = wzbase;
        for (int i = 0; i < 13; ++i) { wz[i] = p; p += (size_t)kpadv[i] * 64; }
    }
    for (int i = 0; i < 13; ++i) {
        const int total = (kpadv[i] >> 2) * 128;
        k_swzW<<<(total + 255) / 256, 256, 0, stream>>>(wsrc[i], kreal[i], kpadv[i], wz[i]);
    }

    // ---------------- Layer 1 ----------------
    k_prep<<<8, 128, 0, stream>>>(positions, features, t_ptr, h1);
    k_scalars1<<<1, 64, 0, stream>>>(positions,
        F(pos1 + 1), F(pos1 + 2), F(pos1 + 3), F(pos1 + 4), F(pos1 + 5),
        F(pos2 + 1), F(pos2 + 2), F(pos2 + 3), F(pos2 + 4), F(pos2 + 5), scal);
    k_buildF1<<<8, 256, 0, stream>>>(h1, positions, Fbuf);

    // msg MLP: 73->64 relu, 64->64 relu, 64->64 (m stored in buf0)
    k_gemm64<<<128, 32, 0, stream>>>(Fbuf, 76, 76,  wz[0],  F(msg1 + 1), buf1, 1);
    k_gemm64<<<128, 32, 0, stream>>>(buf1, 64, 64,  wz[1],  F(msg1 + 3), buf2, 1);
    k_gemm64<<<128, 32, 0, stream>>>(buf2, 64, 64,  wz[2],  F(msg1 + 5), buf0, 0);
    // pos MLP hidden layers on m, then 64->3 head
    k_gemm64<<<128, 32, 0, stream>>>(buf0, 64, 64,  wz[3],  F(pos1 + 1), buf1, 1);
    k_gemm64<<<128, 32, 0, stream>>>(buf1, 64, 64,  wz[4],  F(pos1 + 3), buf2, 1);
    k_pos_last<<<8, 256, 0, stream>>>(buf2, F(pos1 + 4), F(pos1 + 5), pout);

    k_featin<<<4, 256, 0, stream>>>(h1, buf0, Ffeat);
    k_xupd1<<<4, 256, 0, stream>>>(positions, pout, scal, z1);

    // feat MLP: 100->64 relu, 64->64 relu, 64->64 (+outer relu folded)
    k_gemm64<<<64, 32, 0, stream>>>(Ffeat, 100, 100, wz[5],  F(feat1 + 1), buf1, 1);
    k_gemm64<<<64, 32, 0, stream>>>(buf1,  64,  64,  wz[6],  F(feat1 + 3), buf2, 1);
    k_gemm64<<<64, 32, 0, stream>>>(buf2,  64,  64,  wz[7],  F(feat1 + 5), h2,   1);

    // ---------------- Layer 2 ----------------
    k_scalars2<<<1, 32, 0, stream>>>(z1, scal);
    k_buildF2<<<8, 256, 0, stream>>>(h2, z1, Fbuf);

    k_gemm64<<<128, 32, 0, stream>>>(Fbuf, 132, 132, wz[8],  F(msg2 + 1), buf1, 1);
    k_gemm64<<<128, 32, 0, stream>>>(buf1, 64,  64,  wz[9],  F(msg2 + 3), buf2, 1);
    k_gemm64<<<128, 32, 0, stream>>>(buf2, 64,  64,  wz[10], F(msg2 + 5), buf0, 0);

    k_gemm64<<<128, 32, 0, stream>>>(buf0, 64,  64,  wz[11], F(pos2 + 1), buf1, 1);
    k_gemm64<<<128, 32, 0, stream>>>(buf1, 64,  64,  wz[12], F(pos2 + 3), buf2, 1);
    k_pos_last<<<8, 256, 0, stream>>>(buf2, F(pos2 + 4), F(pos2 + 5), pout);

    k_final<<<4, 256, 0, stream>>>(z1, pout, scal, (float*)d_out);
}